// MultiLabelRankingLossWithIndicatorTarget_32418413150578
// MI455X (gfx1250) — compile-verified
//
#include <hip/hip_runtime.h>

typedef __attribute__((ext_vector_type(2))) float v2f;
typedef __attribute__((ext_vector_type(8))) float v8f;

#define BB 256      // batch rows
#define LL 1024     // labels per row
#define KSPLIT 8    // K-dimension split per row group -> 16*8 = 128 waves
#define KSPAN (LL / KSPLIT)   // 128 K per wave
#define WAVES_PER_BLOCK 8

// ws layout: ws[(q*KSPLIT + kslice)*BB + row], q in {0:sum_all, 1:sum_pos, 2:cnt_pos}

__device__ __forceinline__ float fast_sigmoid(float x) {
    // 1 / (1 + e^-x) with native v_exp_f32 + v_rcp_f32 (~1 ulp), no IEEE div sequence
    return __builtin_amdgcn_rcpf(1.0f + __expf(-x));
}

__global__ __launch_bounds__(256)
void mlrl_partial_kernel(const float* __restrict__ logits,
                         const int*   __restrict__ target,
                         float*       __restrict__ ws) {
    const int waveInBlock = threadIdx.x >> 5;
    const int lane        = threadIdx.x & 31;
    const int w           = blockIdx.x * WAVES_PER_BLOCK + waveInBlock; // 0..127
    const int rowGroup    = w / KSPLIT;   // 0..15  (16 rows each)
    const int kslice      = w % KSPLIT;   // 0..7
    const int half        = lane >> 4;    // 0: K={0,1}, 1: K={2,3} of each 4-tile
    const int mrow        = lane & 15;    // M index 0..15 within group

    const int row = rowGroup * 16 + mrow;
    const float* lrow = logits + (size_t)row * LL;
    const int*   trow = target + (size_t)row * LL;

    v8f acc_all = {};   // D[m,n] = sum of sigmoid over this wave's K-slice
    v8f acc_pos = {};   // masked by target==1
    v8f acc_cnt = {};   // count of target==1
    v2f ones = {1.0f, 1.0f};

    const int kbase = kslice * KSPAN + 2 * half;

    #pragma unroll 4
    for (int t = 0; t < KSPAN / 4; ++t) {
        const int k = kbase + t * 4;
        // lanes l and l+16 of a row together cover 16 contiguous bytes
        const float2 x  = *(const float2*)(lrow + k);
        const int2   tg = *(const int2*)(trow + k);

        const float p0 = fast_sigmoid(x.x);
        const float p1 = fast_sigmoid(x.y);

        v2f a_all = {p0, p1};
        v2f a_pos = {tg.x ? p0 : 0.0f, tg.y ? p1 : 0.0f};
        v2f a_cnt = {(float)tg.x, (float)tg.y};

        // D = A(16x4) * ones(4x16) + C  -> row sums broadcast across N
        acc_all = __builtin_amdgcn_wmma_f32_16x16x4_f32(
            false, a_all, false, ones, (short)0, acc_all, false, false);
        acc_pos = __builtin_amdgcn_wmma_f32_16x16x4_f32(
            false, a_pos, false, ones, (short)0, acc_pos, false, false);
        acc_cnt = __builtin_amdgcn_wmma_f32_16x16x4_f32(
            false, a_cnt, false, ones, (short)0, acc_cnt, false, false);
    }

    // D layout: VGPR j, lanes 0-15 -> M=j, N=lane; lanes 16-31 -> M=8+j, N=lane-16.
    // Column N==0 lives in lanes 0 (rows +0..7) and 16 (rows +8..15).
    if (mrow == 0) {
        const int mbase = rowGroup * 16 + half * 8;
        #pragma unroll
        for (int j = 0; j < 8; ++j) {
            const int r = mbase + j;
            ws[(0 * KSPLIT + kslice) * BB + r] = acc_all[j];
            ws[(1 * KSPLIT + kslice) * BB + r] = acc_pos[j];
            ws[(2 * KSPLIT + kslice) * BB + r] = acc_cnt[j];
        }
    }
}

__global__ __launch_bounds__(256)
void mlrl_finalize_kernel(const float* __restrict__ ws, float* __restrict__ out) {
    __shared__ float sh[BB];
    const int b = threadIdx.x;

    float S = 0.0f, Sp = 0.0f, np = 0.0f;
    #pragma unroll
    for (int ks = 0; ks < KSPLIT; ++ks) {  // fixed order -> deterministic
        S  += ws[(0 * KSPLIT + ks) * BB + b];
        Sp += ws[(1 * KSPLIT + ks) * BB + b];
        np += ws[(2 * KSPLIT + ks) * BB + b];
    }

    const float nn  = (float)LL - np;
    const float Sn  = S - Sp;
    // pair_sum = np*nn - nn*Sp + np*Sn  (hinge always active since sigmoid in [0,1])
    const float num = np * nn - nn * Sp + np * Sn;
    const float den = fmaxf(np * nn, 1.0f);
    sh[b] = num / den;   // one real divide total per row; fine
    __syncthreads();

    #pragma unroll
    for (int s = BB / 2; s > 0; s >>= 1) {
        if (b < s) sh[b] += sh[b + s];
        __syncthreads();
    }
    if (b == 0) out[0] = sh[0] / (float)BB;
}

extern "C" void kernel_launch(void* const* d_in, const int* in_sizes, int n_in,
                              void* d_out, int out_size, void* d_ws, size_t ws_size,
                              hipStream_t stream) {
    const float* logits = (const float*)d_in[0];
    const int*   target = (const int*)d_in[1];
    float* ws  = (float*)d_ws;   // needs 3 * KSPLIT * BB * 4 = 24 KB
    float* out = (float*)d_out;

    // 16 row-groups * KSPLIT waves, 8 waves per block -> 16 blocks of 256 threads
    mlrl_partial_kernel<<<(16 * KSPLIT) / WAVES_PER_BLOCK, 256, 0, stream>>>(logits, target, ws);
    mlrl_finalize_kernel<<<1, 256, 0, stream>>>(ws, out);
}